// HeteroSAGEEncoder_15290083574226
// MI455X (gfx1250) — compile-verified
//
#include <hip/hip_runtime.h>

// ---------------------------------------------------------------------------
// CDNA5 WMMA f32 16x16x4 fragment types:
//   A: 16x4 f32  -> 2 VGPRs/lane (float2)
//   B:  4x16 f32 -> 2 VGPRs/lane
//   C/D: 16x16 f32 -> 8 VGPRs/lane
// ---------------------------------------------------------------------------
typedef __attribute__((ext_vector_type(2))) float v2f;
typedef __attribute__((ext_vector_type(8))) float v8f;

#define HDIM 128
#define LSTR (2 * HDIM + 32)   // 288 dwords per k-pair row: halves hit disjoint bank sets

__device__ __forceinline__ void atomAddF(float* p, float v) {
    unsafeAtomicAdd(p, v);   // hardware global_atomic_add_f32
}

// ---------------------------------------------------------------------------
// out[n, 0:128] = (rowScale(n) * A1[n, 0:K]) @ W1^T + A2[n, 0:K] @ W2^T + bias
// Weights are staged into LDS once per block in a paired-k layout:
//   lds[(k>>1)*LSTR + col*2 + (k&1)] = W[col*K + k]
// so a B fragment (v0=B[kk][col], v1=B[kk+1][col]) is a single aligned
// ds_load_b64, with lanes 0-15 (k-pair p) and 16-31 (p+1) on disjoint banks.
// A fragments are prefetched one K-step ahead of their WMMAs.
// One wave computes a 16-row x 128-col strip (8 col tiles, K step = 4).
// ---------------------------------------------------------------------------
template <int K, bool DUAL, bool SCALED>
__global__ __launch_bounds__(256) void gemm_wmma(
        const float* __restrict__ A1, const float* __restrict__ cnt,
        const float* __restrict__ W1, const float* __restrict__ A2,
        const float* __restrict__ W2, const float* __restrict__ bias,
        float* __restrict__ out, int nrows) {
    __shared__ float lds[(DUAL ? 2 : 1) * (K / 2) * LSTR];
    float* lds1 = lds;
    float* lds2 = lds + (K / 2) * LSTR;

    // Stage weights (coalesced global reads, one-time scattered LDS writes).
    for (int idx = threadIdx.x; idx < HDIM * K; idx += blockDim.x) {
        const int j = idx / K;
        const int k = idx - j * K;
        const int a = (k >> 1) * LSTR + j * 2 + (k & 1);
        lds1[a] = W1[idx];
        if constexpr (DUAL) lds2[a] = W2[idx];
    }
    __syncthreads();

    const int lane = threadIdx.x & 31;
    const int wave = threadIdx.x >> 5;
    const int rowTile = blockIdx.x * (blockDim.x >> 5) + wave;
    const int rowBase = rowTile * 16;
    if (rowBase >= nrows) return;              // wave-uniform; after the barrier

    const int half = lane >> 4;                // 0: lanes 0-15, 1: lanes 16-31
    const int l16  = lane & 15;
    const int arow = rowBase + l16;            // A-fragment row for this lane

    float scale = 1.0f;
    if constexpr (SCALED) scale = 1.0f / fmaxf(cnt[arow], 1.0f);

    v8f acc[8];
    #pragma unroll
    for (int t = 0; t < 8; ++t)
        #pragma unroll
        for (int r = 0; r < 8; ++r) acc[t][r] = 0.0f;

    const float* a1p = A1 + (size_t)arow * K;
    const float* a2p = DUAL ? (A2 + (size_t)arow * K) : nullptr;

    // Prime the A-fragment pipeline.
    v2f a1 = *(const v2f*)(a1p + 2 * half);
    v2f a2;
    if constexpr (DUAL) a2 = *(const v2f*)(a2p + 2 * half);

    for (int kb = 0; kb < K; kb += 4) {
        // Prefetch next A fragments (overlaps with this step's WMMAs).
        v2f a1n = a1, a2n;
        if constexpr (DUAL) a2n = a2;
        if (kb + 4 < K) {
            const int kn = kb + 4 + 2 * half;
            a1n = *(const v2f*)(a1p + kn);
            if constexpr (DUAL) a2n = *(const v2f*)(a2p + kn);
        }

        v2f a1s = a1;
        if constexpr (SCALED) { a1s.x *= scale; a1s.y *= scale; }

        const float* lb1 = lds1 + ((kb >> 1) + half) * LSTR + l16 * 2;
        #pragma unroll
        for (int t = 0; t < 8; ++t) {
            const v2f b1 = *(const v2f*)(lb1 + t * 32);
            acc[t] = __builtin_amdgcn_wmma_f32_16x16x4_f32(
                false, a1s, false, b1, (short)0, acc[t], false, false);
        }
        if constexpr (DUAL) {
            const float* lb2 = lds2 + ((kb >> 1) + half) * LSTR + l16 * 2;
            #pragma unroll
            for (int t = 0; t < 8; ++t) {
                const v2f b2 = *(const v2f*)(lb2 + t * 32);
                acc[t] = __builtin_amdgcn_wmma_f32_16x16x4_f32(
                    false, a2, false, b2, (short)0, acc[t], false, false);
            }
        }
        a1 = a1n;
        if constexpr (DUAL) a2 = a2n;
    }

    // D layout: VGPR r -> M = r + 8*half; N = l16 (per 16-col tile).
    #pragma unroll
    for (int t = 0; t < 8; ++t) {
        const int col = t * 16 + l16;
        const float bv = bias[col];
        #pragma unroll
        for (int r = 0; r < 8; ++r) {
            const int row = rowBase + 8 * half + r;
            out[(size_t)row * HDIM + col] = acc[t][r] + bv;
        }
    }
}

// ---------------------------------------------------------------------------
// One wave per edge: agg[dst] += h[src], cnt[dst] += 1  (segment-sum numerator
// and denominator; the divide is folded into the GEMM's A-scale).
// ---------------------------------------------------------------------------
__global__ void scatter_accum(const float* __restrict__ h, const int* __restrict__ src,
                              const int* __restrict__ dst, float* __restrict__ agg,
                              float* __restrict__ cnt, int nedges) {
    const int lane = threadIdx.x & 31;
    const int e = blockIdx.x * (blockDim.x >> 5) + (threadIdx.x >> 5);
    if (e >= nedges) return;
    const int s = src[e];
    const int d = dst[e];
    const float4 v = *(const float4*)(h + (size_t)s * HDIM + lane * 4);
    float* ap = agg + (size_t)d * HDIM + lane * 4;
    atomAddF(ap + 0, v.x);
    atomAddF(ap + 1, v.y);
    atomAddF(ap + 2, v.z);
    atomAddF(ap + 3, v.w);
    if (lane == 0) atomAddF(cnt + d, 1.0f);
}

__global__ void zero_f32(float* __restrict__ p, size_t n) {
    const size_t i = (size_t)blockIdx.x * blockDim.x + threadIdx.x;
    if (i < n) p[i] = 0.0f;
}

// Per-channel sum and sum-of-squares over the node axis. blockDim.x == 128.
__global__ void bn_stats(const float* __restrict__ x, float* __restrict__ stats, int nrows) {
    const int ch = threadIdx.x;
    float s = 0.0f, q = 0.0f;
    for (int n = blockIdx.x; n < nrows; n += gridDim.x) {
        const float v = x[(size_t)n * HDIM + ch];
        s += v;
        q += v * v;
    }
    atomAddF(&stats[ch], s);
    atomAddF(&stats[HDIM + ch], q);
}

// y = relu(g * (x - mu) * rsqrt(var + eps) + b), biased var from stats.
__global__ void bn_apply(const float* __restrict__ x, const float* __restrict__ stats,
                         const float* __restrict__ g, const float* __restrict__ b,
                         float* __restrict__ out, int nrows) {
    const size_t i = (size_t)blockIdx.x * blockDim.x + threadIdx.x;
    const size_t total = (size_t)nrows * HDIM;
    if (i >= total) return;
    const int ch = (int)(i & (HDIM - 1));
    const float invN = 1.0f / (float)nrows;
    const float mu  = stats[ch] * invN;
    const float var = stats[HDIM + ch] * invN - mu * mu;
    const float rs  = rsqrtf(var + 1e-5f);
    const float v = g[ch] * (x[i] - mu) * rs + b[ch];
    out[i] = fmaxf(v, 0.0f);
}

// ---------------------------------------------------------------------------
extern "C" void kernel_launch(void* const* d_in, const int* in_sizes, int n_in,
                              void* d_out_v, int out_size, void* d_ws, size_t ws_size,
                              hipStream_t stream) {
    const float* x_user     = (const float*)d_in[0];
    const float* x_item     = (const float*)d_in[1];
    const float* lin_user_w = (const float*)d_in[2];
    const float* lin_user_b = (const float*)d_in[3];
    const float* lin_item_w = (const float*)d_in[4];
    const float* lin_item_b = (const float*)d_in[5];
    const float* Wl_ui      = (const float*)d_in[6];
    const float* bl_ui      = (const float*)d_in[7];
    const float* Wr_ui      = (const float*)d_in[8];
    const float* Wl_iu      = (const float*)d_in[9];
    const float* bl_iu      = (const float*)d_in[10];
    const float* Wr_iu      = (const float*)d_in[11];
    const float* bn_g_user  = (const float*)d_in[12];
    const float* bn_b_user  = (const float*)d_in[13];
    const float* bn_g_item  = (const float*)d_in[14];
    const float* bn_b_item  = (const float*)d_in[15];
    const int*   edge_ui    = (const int*)d_in[16];
    const int*   edge_iu    = (const int*)d_in[17];
    float* out = (float*)d_out_v;

    const int H  = in_sizes[3];            // 128
    const int Ku = in_sizes[2] / H;        // 64
    const int N  = in_sizes[0] / Ku;       // 100000
    const int E  = in_sizes[16] / 2;       // 600000
    const int L  = in_sizes[7] / H;        // 2
    const size_t NH = (size_t)N * H;

    // Workspace layout (floats): hu | hi | bufA | bufB | cntA | cntB | stats(4H)
    float* hu    = (float*)d_ws;
    float* hi    = hu + NH;
    float* bufA  = hi + NH;                // agg_i, then out_i (in-place GEMM)
    float* bufB  = bufA + NH;              // agg_u, then out_u
    float* cntA  = bufB + NH;
    float* cntB  = cntA + N;
    float* stats = cntB + N;               // [2H] item stats, [2H] user stats

    const int rowTiles   = (N + 15) / 16;
    const int gemmBlocks = (rowTiles + 7) / 8;     // 8 waves / block (256 thr)
    const int scatBlocks = (E + 7) / 8;            // 8 waves / block (256 thr)
    const unsigned ewBlocks = (unsigned)((NH + 255) / 256);

    // Input projections: hu = x_user @ Wu^T + bu ; hi = x_item @ Wi^T + bi
    gemm_wmma<64, false, false><<<gemmBlocks, 256, 0, stream>>>(
        x_user, nullptr, lin_user_w, nullptr, nullptr, lin_user_b, hu, N);
    gemm_wmma<128, false, false><<<gemmBlocks, 256, 0, stream>>>(
        x_item, nullptr, lin_item_w, nullptr, nullptr, lin_item_b, hi, N);

    for (int l = 0; l < L; ++l) {
        // Zero agg buffers + counts + stats (contiguous region).
        const size_t zc = 2 * NH + 2 * (size_t)N + 4 * (size_t)H;
        zero_f32<<<(unsigned)((zc + 255) / 256), 256, 0, stream>>>(bufA, zc);

        // Segment sums (user->item and item->user).
        scatter_accum<<<scatBlocks, 256, 0, stream>>>(hu, edge_ui, edge_ui + E, bufA, cntA, E);
        scatter_accum<<<scatBlocks, 256, 0, stream>>>(hi, edge_iu, edge_iu + E, bufB, cntB, E);

        // out_i = mean(agg_i) @ Wl_ui^T + hi @ Wr_ui^T + bl_ui   (in-place bufA)
        gemm_wmma<128, true, true><<<gemmBlocks, 256, 0, stream>>>(
            bufA, cntA, Wl_ui + (size_t)l * H * H, hi, Wr_ui + (size_t)l * H * H,
            bl_ui + (size_t)l * H, bufA, N);
        // out_u = mean(agg_u) @ Wl_iu^T + hu @ Wr_iu^T + bl_iu   (in-place bufB)
        gemm_wmma<128, true, true><<<gemmBlocks, 256, 0, stream>>>(
            bufB, cntB, Wl_iu + (size_t)l * H * H, hu, Wr_iu + (size_t)l * H * H,
            bl_iu + (size_t)l * H, bufB, N);

        // BatchNorm stats over node axis.
        bn_stats<<<512, 128, 0, stream>>>(bufA, stats, N);
        bn_stats<<<512, 128, 0, stream>>>(bufB, stats + 2 * H, N);

        // hi = relu(bn(out_i, item)) ; hu = relu(bn(out_u, user)).
        float* hi_dst = (l == L - 1) ? (out + NH) : hi;
        float* hu_dst = (l == L - 1) ? out : hu;
        bn_apply<<<ewBlocks, 256, 0, stream>>>(
            bufA, stats, bn_g_item + (size_t)l * H, bn_b_item + (size_t)l * H, hi_dst, N);
        bn_apply<<<ewBlocks, 256, 0, stream>>>(
            bufB, stats + 2 * H, bn_g_user + (size_t)l * H, bn_b_user + (size_t)l * H, hu_dst, N);
    }
}